// SimplicialModel_18932215841010
// MI455X (gfx1250) — compile-verified
//
#include <hip/hip_runtime.h>
#include <hip/hip_bf16.h>
#include <math.h>

typedef __attribute__((ext_vector_type(2))) float v2f;
typedef __attribute__((ext_vector_type(8))) float v8f;

// D = A(16x4,f32) * B(4x16,f32) + C(16x16,f32)   -- V_WMMA_F32_16X16X4_F32
__device__ __forceinline__ v8f wmma4(v2f a, v2f b, v8f c) {
    return __builtin_amdgcn_wmma_f32_16x16x4_f32(false, a, false, b, (short)0, c,
                                                 false, false);
}

// Problem constants (reference: B=16, S=512, D=256, H=1, dh=64, ve=64)
#define NB   16
#define SEQ  512
#define DIM  256
#define DH   64
#define VE   64
#define SQ   448   // SEQ - VE

// ---------------------------------------------------------------------------
// Kernel 1: qkkv = x @ W      (8192 x 256) = (8192 x 256)(256 x 256)
// block (32,4): wave -> ntile, grid (4, 512)
// ---------------------------------------------------------------------------
__global__ void k_qkkv(const float* __restrict__ x, const float* __restrict__ w,
                       float* __restrict__ qkkv) {
    const int lane = threadIdx.x, half = lane >> 4, mn = lane & 15, kk0 = half * 2;
    const int ntile = blockIdx.x * blockDim.y + threadIdx.y;  // 0..15
    const int mtile = blockIdx.y;                             // 0..511
    const int mrow = mtile * 16 + mn;
    const int ncol = ntile * 16 + mn;
    v8f c = {};
    for (int k = 0; k < DIM; k += 4) {
        v2f a, b;
        a.x = x[(size_t)mrow * DIM + k + kk0];
        a.y = x[(size_t)mrow * DIM + k + kk0 + 1];
        b.x = w[(size_t)(k + kk0) * DIM + ncol];
        b.y = w[(size_t)(k + kk0 + 1) * DIM + ncol];
        c = wmma4(a, b, c);
    }
    for (int g = 0; g < 8; ++g)
        qkkv[(size_t)(mtile * 16 + g + 8 * half) * DIM + ncol] = c[g];
}

// ---------------------------------------------------------------------------
// Kernel 2: qk1[a,i,j] = q.k1^T, qk2[a,i,k] = q.k2^T   (448x64, K=64 per batch)
// block (32,8): waves 0-3 -> qk1 ntiles, 4-7 -> qk2 ntiles; grid (1,28,16)
// ---------------------------------------------------------------------------
__global__ void k_qk12(const float* __restrict__ qkkv, float* __restrict__ qk1,
                       float* __restrict__ qk2) {
    const int lane = threadIdx.x, half = lane >> 4, mn = lane & 15, kk0 = half * 2;
    const int w = threadIdx.y, which = w >> 2, ntile = w & 3;
    const int itile = blockIdx.y, a = blockIdx.z;
    const float* qb = qkkv + (size_t)(a * SEQ) * DIM;                       // q rows, cols 0..63
    const float* kb = qkkv + (size_t)(a * SEQ + SQ) * DIM + 64 + which * 64; // k1 or k2
    const int m = itile * 16 + mn, n = ntile * 16 + mn;
    v8f c = {};
    for (int k = 0; k < DH; k += 4) {
        v2f av, bv;
        av.x = qb[(size_t)m * DIM + k + kk0];
        av.y = qb[(size_t)m * DIM + k + kk0 + 1];
        bv.x = kb[(size_t)n * DIM + k + kk0];      // B[k][n] = K[n][k]
        bv.y = kb[(size_t)n * DIM + k + kk0 + 1];
        c = wmma4(av, bv, c);
    }
    float* dst = which ? qk2 : qk1;
    for (int g = 0; g < 8; ++g) {
        int i = itile * 16 + g + 8 * half;
        dst[(size_t)(a * SQ + i) * DH + ntile * 16 + mn] = c[g];
    }
}

// ---------------------------------------------------------------------------
// Kernel 3: k1k2[a,j,k] = k1.k2^T   (64x64, K=64) ; block (32,16), grid 16
// ---------------------------------------------------------------------------
__global__ void k_k1k2(const float* __restrict__ qkkv, float* __restrict__ k1k2) {
    const int lane = threadIdx.x, half = lane >> 4, mn = lane & 15, kk0 = half * 2;
    const int w = threadIdx.y, jt = w >> 2, kt = w & 3;
    const int a = blockIdx.x;
    const float* k1b = qkkv + (size_t)(a * SEQ + SQ) * DIM + 64;
    const float* k2b = qkkv + (size_t)(a * SEQ + SQ) * DIM + 128;
    const int m = jt * 16 + mn, n = kt * 16 + mn;
    v8f c = {};
    for (int k = 0; k < DH; k += 4) {
        v2f av, bv;
        av.x = k1b[(size_t)m * DIM + k + kk0];
        av.y = k1b[(size_t)m * DIM + k + kk0 + 1];
        bv.x = k2b[(size_t)n * DIM + k + kk0];
        bv.y = k2b[(size_t)n * DIM + k + kk0 + 1];
        c = wmma4(av, bv, c);
    }
    for (int g = 0; g < 8; ++g)
        k1k2[(size_t)(a * DH + jt * 16 + g + 8 * half) * DH + kt * 16 + mn] = c[g];
}

// ---------------------------------------------------------------------------
// Kernel 4: row norms qq (16x448), k1k1 (16x64), k2k2 (16x64)
// ---------------------------------------------------------------------------
__global__ void k_norms(const float* __restrict__ qkkv, float* __restrict__ qq,
                        float* __restrict__ k1k1, float* __restrict__ k2k2) {
    int idx = blockIdx.x * blockDim.x + threadIdx.x;
    if (idx >= NB * 576) return;
    int a = idx / 576, r = idx % 576;
    const float* base; float* dst;
    if (r < SQ)      { base = qkkv + (size_t)(a * SEQ + r) * DIM;                 dst = qq   + a * SQ + r; }
    else if (r < 512){ base = qkkv + (size_t)(a * SEQ + SQ + (r - SQ)) * DIM + 64; dst = k1k1 + a * DH + (r - SQ); }
    else             { base = qkkv + (size_t)(a * SEQ + SQ + (r - 512)) * DIM + 128; dst = k2k2 + a * DH + (r - 512); }
    float s = 0.f;
    for (int d = 0; d < DH; ++d) { float t = base[d]; s += t * t; }
    *dst = s;
}

// ---------------------------------------------------------------------------
// Kernel 5a: tmp[a,j,(q*64+s)] = sum_r v[a,j,r] * Bw[q,r,s]   M=64,N=4096,K=64
// block (32,4): wave -> mtile; grid (256, 16) = (ntile, a)
// ---------------------------------------------------------------------------
__global__ void k_vb(const float* __restrict__ qkkv, const float* __restrict__ Bw,
                     float* __restrict__ tmp) {
    const int lane = threadIdx.x, half = lane >> 4, mn = lane & 15, kk0 = half * 2;
    const int mtile = threadIdx.y, ntile = blockIdx.x, a = blockIdx.y;
    const float* vb = qkkv + (size_t)(a * SEQ + SQ) * DIM + 192;  // v[a]
    const int m = mtile * 16 + mn;
    const int n = ntile * 16 + mn;
    const int qi = n >> 6, si = n & 63;
    v8f c = {};
    for (int k = 0; k < DH; k += 4) {
        v2f av, bv;
        av.x = vb[(size_t)m * DIM + k + kk0];
        av.y = vb[(size_t)m * DIM + k + kk0 + 1];
        bv.x = Bw[(size_t)qi * 4096 + (k + kk0) * 64 + si];     // B[r][(q,s)]
        bv.y = Bw[(size_t)qi * 4096 + (k + kk0 + 1) * 64 + si];
        c = wmma4(av, bv, c);
    }
    for (int g = 0; g < 8; ++g)
        tmp[(size_t)(a * DH + mtile * 16 + g + 8 * half) * 4096 + n] = c[g];
}

// ---------------------------------------------------------------------------
// Kernel 5b: Bv[a,q,j,k] = sum_s tmp[a,j,q*64+s] * v[a,k,s]   M=64,N=64,K=64
// block (32,16); grid (64, 16) = (q, a)
// ---------------------------------------------------------------------------
__global__ void k_bv(const float* __restrict__ qkkv, const float* __restrict__ tmp,
                     float* __restrict__ bv) {
    const int lane = threadIdx.x, half = lane >> 4, mn = lane & 15, kk0 = half * 2;
    const int w = threadIdx.y, jt = w >> 2, kt = w & 3;
    const int qi = blockIdx.x, a = blockIdx.y;
    const float* vb = qkkv + (size_t)(a * SEQ + SQ) * DIM + 192;
    const int m = jt * 16 + mn, n = kt * 16 + mn;
    v8f c = {};
    for (int s = 0; s < DH; s += 4) {
        v2f av, bvv;
        av.x = tmp[(size_t)(a * DH + m) * 4096 + qi * 64 + s + kk0];
        av.y = tmp[(size_t)(a * DH + m) * 4096 + qi * 64 + s + kk0 + 1];
        bvv.x = vb[(size_t)n * DIM + s + kk0];      // B[s][k] = v[k][s]
        bvv.y = vb[(size_t)n * DIM + s + kk0 + 1];
        c = wmma4(av, bvv, c);
    }
    for (int g = 0; g < 8; ++g)
        bv[((size_t)(a * DH + qi) * DH + jt * 16 + g + 8 * half) * DH + kt * 16 + mn] = c[g];
}

// ---------------------------------------------------------------------------
// Kernel 6: per-row softmax stats: m[a,i]=max logits, Z[a,i]=sum exp(l-m)
// one wave per row; block (32,8); grid 896
// ---------------------------------------------------------------------------
__device__ __forceinline__ float logit_val(float A1, float A2, float C,
                                           float qqv, float n1, float n2) {
    float pre = A1 * A1 * n2 + C * C * qqv + A2 * A2 * n1 - 2.f * A1 * A2 * C;
    return sqrtf(fmaxf(pre, 0.f));
}

__global__ void k_stats(const float* __restrict__ qk1, const float* __restrict__ qk2,
                        const float* __restrict__ k1k2, const float* __restrict__ qq,
                        const float* __restrict__ k1k1, const float* __restrict__ k2k2,
                        float* __restrict__ mbuf, float* __restrict__ zbuf) {
    const int lane = threadIdx.x;
    const int row = blockIdx.x * blockDim.y + threadIdx.y;  // 0..7167
    if (row >= NB * SQ) return;
    const int a = row / SQ;
    const float* q1r = qk1 + (size_t)row * DH;
    const float* q2r = qk2 + (size_t)row * DH;
    const float* kkb = k1k2 + (size_t)a * DH * DH;
    const float* n1b = k1k1 + a * DH;
    const float* n2b = k2k2 + a * DH;
    const float qqv = qq[row];
    float mx = 0.f;  // logits are >= 0
    for (int t = lane; t < DH * DH; t += 32) {
        int j = t >> 6, k = t & 63;
        mx = fmaxf(mx, logit_val(q1r[j], q2r[k], kkb[t], qqv, n1b[j], n2b[k]));
    }
    for (int o = 16; o > 0; o >>= 1) mx = fmaxf(mx, __shfl_xor(mx, o, 32));
    float sum = 0.f;
    for (int t = lane; t < DH * DH; t += 32) {
        int j = t >> 6, k = t & 63;
        sum += __expf(logit_val(q1r[j], q2r[k], kkb[t], qqv, n1b[j], n2b[k]) - mx);
    }
    for (int o = 16; o > 0; o >>= 1) sum += __shfl_xor(sum, o, 32);
    if (lane == 0) { mbuf[row] = mx; zbuf[row] = sum; }
}

// ---------------------------------------------------------------------------
// Kernel 7: heads[a,i,q] = (1/Z_i) sum_jk e[i,jk] * Bv[a,q,jk]
// e tiles (16x256) recomputed into LDS, consumed as WMMA A-operand.
// block (32,4) -> 4 n-tiles; grid (28, 16) = (itile, a)
// ---------------------------------------------------------------------------
__global__ void k_heads(const float* __restrict__ qk1, const float* __restrict__ qk2,
                        const float* __restrict__ k1k2, const float* __restrict__ qq,
                        const float* __restrict__ k1k1, const float* __restrict__ k2k2,
                        const float* __restrict__ mbuf, const float* __restrict__ zbuf,
                        const float* __restrict__ bv, float* __restrict__ out) {
    __shared__ float e[16 * 256];
    __shared__ float mrow[16];
    const int lane = threadIdx.x, half = lane >> 4, mn = lane & 15, kk0 = half * 2;
    const int tid = threadIdx.y * 32 + lane;
    const int itile = blockIdx.x, a = blockIdx.y;
    if (tid < 16) mrow[tid] = mbuf[a * SQ + itile * 16 + tid];
    const float* kkb = k1k2 + (size_t)a * DH * DH;
    const float* n1b = k1k1 + a * DH;
    const float* n2b = k2k2 + a * DH;
    const float* bvb = bv + (size_t)a * DH * 4096;
    const int n = threadIdx.y * 16 + mn;
    v8f c = {};
    for (int ch = 0; ch < 16; ++ch) {
        __syncthreads();
        for (int t = tid; t < 16 * 256; t += 128) {
            int il = t >> 8, jkl = t & 255;
            int jk = ch * 256 + jkl;
            int j = jk >> 6, k = jk & 63;
            int row = a * SQ + itile * 16 + il;
            float l = logit_val(qk1[(size_t)row * DH + j], qk2[(size_t)row * DH + k],
                                kkb[jk], qq[row], n1b[j], n2b[k]);
            e[il * 256 + jkl] = __expf(l - mrow[il]);
        }
        __syncthreads();
        for (int k = 0; k < 256; k += 4) {
            v2f av, bvv;
            av.x = e[mn * 256 + k + kk0];
            av.y = e[mn * 256 + k + kk0 + 1];
            bvv.x = bvb[(size_t)n * 4096 + ch * 256 + k + kk0];
            bvv.y = bvb[(size_t)n * 4096 + ch * 256 + k + kk0 + 1];
            c = wmma4(av, bvv, c);
        }
    }
    for (int g = 0; g < 8; ++g) {
        int i = itile * 16 + g + 8 * half;
        float z = zbuf[a * SQ + i];
        out[((size_t)a * SEQ + i) * DH + n] = c[g] / z;
    }
}

// ---------------------------------------------------------------------------
// Kernel 8: tail copy — out rows s in [448,512) = v (virtual) rows
// ---------------------------------------------------------------------------
__global__ void k_tail(const float* __restrict__ qkkv, float* __restrict__ out) {
    int idx = blockIdx.x * blockDim.x + threadIdx.x;  // 0..65535
    if (idx >= NB * VE * DH) return;
    int a = idx >> 12, r = (idx >> 6) & 63, d = idx & 63;
    out[((size_t)a * SEQ + SQ + r) * DH + d] =
        qkkv[(size_t)(a * SEQ + SQ + r) * DIM + 192 + d];
}

// ---------------------------------------------------------------------------
extern "C" void kernel_launch(void* const* d_in, const int* in_sizes, int n_in,
                              void* d_out, int out_size, void* d_ws, size_t ws_size,
                              hipStream_t stream) {
    (void)in_sizes; (void)n_in; (void)out_size; (void)ws_size;
    const float* x  = (const float*)d_in[0];  // (16,512,256)
    const float* w  = (const float*)d_in[1];  // (256,256)
    const float* Bw = (const float*)d_in[2];  // (64,64,64)
    float* out = (float*)d_out;               // (16,512,64)

    float* ws = (float*)d_ws;
    float* qkkv = ws;                     // 2,097,152
    float* qk1  = qkkv + 2097152;         //   458,752
    float* qk2  = qk1  + 458752;          //   458,752
    float* k1k2 = qk2  + 458752;          //    65,536
    float* qq   = k1k2 + 65536;           //     7,168
    float* k1k1 = qq   + 7168;            //     1,024
    float* k2k2 = k1k1 + 1024;            //     1,024
    float* mbuf = k2k2 + 1024;            //     7,168
    float* zbuf = mbuf + 7168;            //     7,168
    float* tmp  = zbuf + 7168;            // 4,194,304
    float* bv   = tmp  + 4194304;         // 4,194,304  (total ~44 MiB)

    k_qkkv <<<dim3(4, 512),  dim3(32, 4),  0, stream>>>(x, w, qkkv);
    k_qk12 <<<dim3(1, 28, 16), dim3(32, 8), 0, stream>>>(qkkv, qk1, qk2);
    k_k1k2 <<<dim3(16),      dim3(32, 16), 0, stream>>>(qkkv, k1k2);
    k_norms<<<dim3(36),      dim3(256),    0, stream>>>(qkkv, qq, k1k1, k2k2);
    k_vb   <<<dim3(256, 16), dim3(32, 4),  0, stream>>>(qkkv, Bw, tmp);
    k_bv   <<<dim3(64, 16),  dim3(32, 16), 0, stream>>>(qkkv, tmp, bv);
    k_stats<<<dim3(896),     dim3(32, 8),  0, stream>>>(qk1, qk2, k1k2, qq, k1k1, k2k2,
                                                        mbuf, zbuf);
    k_heads<<<dim3(28, 16),  dim3(32, 4),  0, stream>>>(qk1, qk2, k1k2, qq, k1k1, k2k2,
                                                        mbuf, zbuf, bv, out);
    k_tail <<<dim3(256),     dim3(256),    0, stream>>>(qkkv, out);
}